// IAR_73031623901810
// MI455X (gfx1250) — compile-verified
//
#include <hip/hip_runtime.h>
#include <hip/hip_bf16.h>
#include <math.h>

typedef __attribute__((ext_vector_type(2))) float v2f;
typedef __attribute__((ext_vector_type(8))) float v8f;

#define N_NODES   100000
#define NNZ_E     1600000
#define CF_B      8192
#define D_ALL     352   // 128+128+64+32
#define D_SLAB    224   // 128+64+32 (layer outputs; layer-0 part read from embed_table)
#define LAMBDA    1e-5f

// ---------------------------------------------------------------------------
// Repack W[din,dout] (row-major) into paired-K layout:
//   Wp[(k/2)*dout + n] = float2{ W[k][n], W[k+1][n] }
// so each WMMA B-fragment is a single aligned b64 load.
// ---------------------------------------------------------------------------
__global__ __launch_bounds__(256) void repack_kernel(
    const float* __restrict__ W, float* __restrict__ Wp, int din, int dout)
{
    int gid = blockIdx.x * blockDim.x + threadIdx.x;
    int total = (din >> 1) * dout;
    if (gid >= total) return;
    int pr = gid / dout;
    int n  = gid - pr * dout;
    v2f v;
    v.x = W[(size_t)(2 * pr) * dout + n];
    v.y = W[(size_t)(2 * pr + 1) * dout + n];
    ((v2f*)Wp)[gid] = v;
}

// ---------------------------------------------------------------------------
// SpMM (COO): side[row] += val * ego[col]; one thread per (edge, 4-float chunk)
// ---------------------------------------------------------------------------
__global__ __launch_bounds__(256) void spmm_kernel(
    const int* __restrict__ rows, const int* __restrict__ cols,
    const float* __restrict__ vals, const float* __restrict__ ego,
    float* __restrict__ side, int nnz, int qshift /* chunks = 1<<qshift, din = 4<<qshift */)
{
    unsigned long long gid = (unsigned long long)blockIdx.x * blockDim.x + threadIdx.x;
    unsigned long long total = (unsigned long long)nnz << qshift;
    if (gid >= total) return;
    int e = (int)(gid >> qshift);
    int q = (int)(gid & ((1u << qshift) - 1u));
    int r = rows[e];
    int c = cols[e];
    float v = vals[e];
    int din = 4 << qshift;
    const float4* src = (const float4*)(ego + (size_t)c * din);
    float4 x = src[q];
    float* dst = side + (size_t)r * din + (q << 2);
    atomicAdd(dst + 0, v * x.x);
    atomicAdd(dst + 1, v * x.y);
    atomicAdd(dst + 2, v * x.z);
    atomicAdd(dst + 3, v * x.w);
}

// ---------------------------------------------------------------------------
// Bi-interaction layer with FP32 WMMA (V_WMMA_F32_16X16X4_F32):
//   out = leaky((ego+side)@W1 + b1) + leaky((ego*side)@W2 + b2)
//   ego_out = out (raw, feeds next layer); slab[...] = l2_normalize(out)
// One wave owns a full 16-row x DOUT output strip so the row L2-norm is a
// local 16-lane butterfly reduction. Wp1/Wp2 are paired-K repacked weights.
// ---------------------------------------------------------------------------
template<int DIN, int DOUT>
__global__ __launch_bounds__(256) void bi_gemm_kernel(
    const float* __restrict__ ego, const float* __restrict__ side,
    const float* __restrict__ Wp1, const float* __restrict__ b1,
    const float* __restrict__ Wp2, const float* __restrict__ b2,
    float* __restrict__ ego_out, float* __restrict__ slab,
    int n_rows, int slab_stride, int slab_off)
{
    constexpr int NT = DOUT / 16;           // column tiles
    const int lane = threadIdx.x & 31;
    const int wave = threadIdx.x >> 5;
    const int tile = blockIdx.x * (blockDim.x >> 5) + wave;
    if (tile * 16 >= n_rows) return;        // whole-wave exit: EXEC stays all-ones
    const int row0 = tile * 16;
    const int m = lane & 15;                // A row / B column within tile
    const int h = lane >> 4;                // half-wave selects K pair

    v8f acc1[NT], acc2[NT];
    for (int t = 0; t < NT; ++t) { acc1[t] = {}; acc2[t] = {}; }

    const float* egoR  = ego  + (size_t)(row0 + m) * DIN;
    const float* sideR = side + (size_t)(row0 + m) * DIN;
    const v2f* W1v = (const v2f*)Wp1;
    const v2f* W2v = (const v2f*)Wp2;

    for (int k0 = 0; k0 < DIN; k0 += 4) {
        const int kk = k0 + 2 * h;          // this lane's K (j=0), kk+1 (j=1)
        v2f e = *(const v2f*)(egoR  + kk);
        v2f s = *(const v2f*)(sideR + kk);
        v2f a1; a1.x = e.x + s.x; a1.y = e.y + s.y;   // (ego+side) fragment
        v2f a2; a2.x = e.x * s.x; a2.y = e.y * s.y;   // (ego*side) fragment
        const size_t pbase = (size_t)((k0 >> 1) + h) * DOUT;
        for (int t = 0; t < NT; ++t) {
            const int n = t * 16 + m;
            v2f bb1 = W1v[pbase + n];       // {W1[kk][n], W1[kk+1][n]} : one b64 load
            v2f bb2 = W2v[pbase + n];
            acc1[t] = __builtin_amdgcn_wmma_f32_16x16x4_f32(
                false, a1, false, bb1, (short)0, acc1[t], false, false);
            acc2[t] = __builtin_amdgcn_wmma_f32_16x16x4_f32(
                false, a2, false, bb2, (short)0, acc2[t], false, false);
        }
    }

    // Epilogue: bias + leaky_relu(0.01) + combine; raw store; accumulate row sumsq.
    float rowsq[8];
    for (int r = 0; r < 8; ++r) rowsq[r] = 0.0f;
    for (int t = 0; t < NT; ++t) {
        const int n = t * 16 + m;
        const float bias1 = b1[n];
        const float bias2 = b2[n];
        for (int r = 0; r < 8; ++r) {
            float v1 = acc1[t][r] + bias1; v1 = (v1 > 0.0f) ? v1 : 0.01f * v1;
            float v2 = acc2[t][r] + bias2; v2 = (v2 > 0.0f) ? v2 : 0.01f * v2;
            float v = v1 + v2;
            acc1[t][r] = v;                  // reuse acc1 as output storage
            rowsq[r] += v * v;
            ego_out[(size_t)(row0 + r + 8 * h) * DOUT + n] = v;
        }
    }
    // Row-wise L2 norm: reduce across the 16 lanes of each half-wave.
    for (int r = 0; r < 8; ++r) {
        float sq = rowsq[r];
        sq += __shfl_xor(sq, 1, 32);
        sq += __shfl_xor(sq, 2, 32);
        sq += __shfl_xor(sq, 4, 32);
        sq += __shfl_xor(sq, 8, 32);
        const float inv = 1.0f / fmaxf(sqrtf(sq), 1e-12f);
        const size_t rowbase = (size_t)(row0 + r + 8 * h) * slab_stride + slab_off;
        for (int t = 0; t < NT; ++t)
            slab[rowbase + t * 16 + m] = acc1[t][r] * inv;
    }
}

// ---------------------------------------------------------------------------
// Per-sample BPR + L2 term; one wave per sample over the 352-dim concat row.
// ---------------------------------------------------------------------------
__global__ __launch_bounds__(256) void sample_kernel(
    const float* __restrict__ embed, const float* __restrict__ slab,
    const int* __restrict__ uid, const int* __restrict__ pid,
    const int* __restrict__ nid, float* __restrict__ samp, int B)
{
    const int lane = threadIdx.x & 31;
    const int wave = threadIdx.x >> 5;
    const int s = blockIdx.x * (blockDim.x >> 5) + wave;
    if (s >= B) return;
    const int u = uid[s], p = pid[s], ng = nid[s];
    float dp = 0.f, dn = 0.f, su = 0.f, sp = 0.f, sn = 0.f;
    for (int c = lane; c < D_ALL; c += 32) {
        float uv, pv, nv;
        if (c < 128) {
            uv = embed[(size_t)u  * 128 + c];
            pv = embed[(size_t)p  * 128 + c];
            nv = embed[(size_t)ng * 128 + c];
        } else {
            uv = slab[(size_t)u  * D_SLAB + (c - 128)];
            pv = slab[(size_t)p  * D_SLAB + (c - 128)];
            nv = slab[(size_t)ng * D_SLAB + (c - 128)];
        }
        dp += uv * pv; dn += uv * nv;
        su += uv * uv; sp += pv * pv; sn += nv * nv;
    }
    for (int mask = 1; mask < 32; mask <<= 1) {
        dp += __shfl_xor(dp, mask, 32);
        dn += __shfl_xor(dn, mask, 32);
        su += __shfl_xor(su, mask, 32);
        sp += __shfl_xor(sp, mask, 32);
        sn += __shfl_xor(sn, mask, 32);
    }
    if (lane == 0) {
        const float x = dp - dn;
        // -log_sigmoid(x) = softplus(-x), numerically stable
        const float loss = fmaxf(-x, 0.0f) + log1pf(expf(-fabsf(x)));
        samp[s] = loss + LAMBDA * 0.5f * (su + sp + sn);
    }
}

__global__ __launch_bounds__(256) void reduce_kernel(
    const float* __restrict__ samp, float* __restrict__ out, int B)
{
    __shared__ float sd[256];
    float s = 0.f;
    for (int i = threadIdx.x; i < B; i += 256) s += samp[i];  // fixed order
    sd[threadIdx.x] = s;
    __syncthreads();
    for (int w = 128; w > 0; w >>= 1) {
        if (threadIdx.x < w) sd[threadIdx.x] += sd[threadIdx.x + w];
        __syncthreads();
    }
    if (threadIdx.x == 0) out[0] = sd[0] / (float)B;
}

// ---------------------------------------------------------------------------
extern "C" void kernel_launch(void* const* d_in, const int* in_sizes, int n_in,
                              void* d_out, int out_size, void* d_ws, size_t ws_size,
                              hipStream_t stream) {
    const float* embed = (const float*)d_in[0];
    const float* W1_0 = (const float*)d_in[1];
    const float* b1_0 = (const float*)d_in[2];
    const float* W2_0 = (const float*)d_in[3];
    const float* b2_0 = (const float*)d_in[4];
    const float* W1_1 = (const float*)d_in[5];
    const float* b1_1 = (const float*)d_in[6];
    const float* W2_1 = (const float*)d_in[7];
    const float* b2_1 = (const float*)d_in[8];
    const float* W1_2 = (const float*)d_in[9];
    const float* b1_2 = (const float*)d_in[10];
    const float* W2_2 = (const float*)d_in[11];
    const float* b2_2 = (const float*)d_in[12];
    const int*   erow = (const int*)d_in[13];
    const int*   ecol = (const int*)d_in[14];
    const float* eval_ = (const float*)d_in[15];
    const int*   uid  = (const int*)d_in[16];
    const int*   pid  = (const int*)d_in[17];
    const int*   nid  = (const int*)d_in[18];

    const int N = N_NODES;
    float* ws    = (float*)d_ws;
    float* side  = ws;                            // N*128
    float* egoA  = side  + (size_t)N * 128;       // N*128
    float* egoB  = egoA  + (size_t)N * 128;       // N*64
    float* slab  = egoB  + (size_t)N * 64;        // N*224
    float* samp  = slab  + (size_t)N * D_SLAB;    // CF_B
    float* wp1_0 = samp  + CF_B;                  // 128*128
    float* wp2_0 = wp1_0 + 128 * 128;
    float* wp1_1 = wp2_0 + 128 * 128;             // 128*64
    float* wp2_1 = wp1_1 + 128 * 64;
    float* wp1_2 = wp2_1 + 128 * 64;              // 64*32
    float* wp2_2 = wp1_2 + 64 * 32;

    // ---- one-time weight repack into paired-K layout ----
    repack_kernel<<<(128 / 2 * 128 + 255) / 256, 256, 0, stream>>>(W1_0, wp1_0, 128, 128);
    repack_kernel<<<(128 / 2 * 128 + 255) / 256, 256, 0, stream>>>(W2_0, wp2_0, 128, 128);
    repack_kernel<<<(128 / 2 * 64 + 255) / 256, 256, 0, stream>>>(W1_1, wp1_1, 128, 64);
    repack_kernel<<<(128 / 2 * 64 + 255) / 256, 256, 0, stream>>>(W2_1, wp2_1, 128, 64);
    repack_kernel<<<(64 / 2 * 32 + 255) / 256, 256, 0, stream>>>(W1_2, wp1_2, 64, 32);
    repack_kernel<<<(64 / 2 * 32 + 255) / 256, 256, 0, stream>>>(W2_2, wp2_2, 64, 32);

    const int tiles = N / 16;                    // 6250
    const int gemm_blocks = (tiles + 7) / 8;     // 8 waves/block

    // ---- layer 0: 128 -> 128 (ego_in = embed_table) ----
    hipMemsetAsync(side, 0, (size_t)N * 128 * sizeof(float), stream);
    {
        unsigned long long tot = (unsigned long long)NNZ_E << 5;   // chunks=32
        int blocks = (int)((tot + 255) / 256);
        spmm_kernel<<<blocks, 256, 0, stream>>>(erow, ecol, eval_, embed, side, NNZ_E, 5);
    }
    bi_gemm_kernel<128, 128><<<gemm_blocks, 256, 0, stream>>>(
        embed, side, wp1_0, b1_0, wp2_0, b2_0, egoA, slab, N, D_SLAB, 0);

    // ---- layer 1: 128 -> 64 ----
    hipMemsetAsync(side, 0, (size_t)N * 128 * sizeof(float), stream);
    {
        unsigned long long tot = (unsigned long long)NNZ_E << 5;
        int blocks = (int)((tot + 255) / 256);
        spmm_kernel<<<blocks, 256, 0, stream>>>(erow, ecol, eval_, egoA, side, NNZ_E, 5);
    }
    bi_gemm_kernel<128, 64><<<gemm_blocks, 256, 0, stream>>>(
        egoA, side, wp1_1, b1_1, wp2_1, b2_1, egoB, slab, N, D_SLAB, 128);

    // ---- layer 2: 64 -> 32 ----
    hipMemsetAsync(side, 0, (size_t)N * 64 * sizeof(float), stream);
    {
        unsigned long long tot = (unsigned long long)NNZ_E << 4;   // chunks=16
        int blocks = (int)((tot + 255) / 256);
        spmm_kernel<<<blocks, 256, 0, stream>>>(erow, ecol, eval_, egoB, side, NNZ_E, 4);
    }
    bi_gemm_kernel<64, 32><<<gemm_blocks, 256, 0, stream>>>(
        egoB, side, wp1_2, b1_2, wp2_2, b2_2, egoA, slab, N, D_SLAB, 192);

    // ---- BPR loss + L2 over 8192 triples ----
    sample_kernel<<<CF_B / 8, 256, 0, stream>>>(embed, slab, uid, pid, nid, samp, CF_B);
    reduce_kernel<<<1, 256, 0, stream>>>(samp, (float*)d_out, CF_B);
}